// GCN_82111184764947
// MI455X (gfx1250) — compile-verified
//
#include <hip/hip_runtime.h>

// ---------------------------------------------------------------------------
// GCN (5-layer, improved self-loops) for MI455X / gfx1250, wave32 + WMMA.
//   Layer: t = h @ W ;  h' = bias + 2*dinv^2 * t  +  sum_e dinv[s]*dinv[d]*t[s]
//   ReLU fused into next layer's GEMM A-load. Final layer has no ReLU.
// ---------------------------------------------------------------------------

#define GCN_N 50000
#define GCN_E 800000
#define GCN_IN 128
#define GCN_H 64

// LDS B-staging: k-paired layout, padded row stride (in floats) so that
// lanes 16..31 (pair-row +1) hit different banks than lanes 0..15.
#define PAIR_STRIDE 130

typedef float v2f __attribute__((ext_vector_type(2)));
typedef float v8f __attribute__((ext_vector_type(8)));

// ---------------- degree / norm precompute --------------------------------

__global__ void gcn_deg_init(float* __restrict__ deg, int n) {
    int i = blockIdx.x * blockDim.x + threadIdx.x;
    if (i < n) deg[i] = 2.0f;                      // self-loop weight = 2
}

__global__ void gcn_deg_edges(const int* __restrict__ ei, float* __restrict__ deg, int e) {
    int i = blockIdx.x * blockDim.x + threadIdx.x;
    if (i < e) atomicAdd(&deg[ei[GCN_E + i]], 1.0f);   // dst row
}

__global__ void gcn_dinv(const float* __restrict__ deg, float* __restrict__ dinv, int n) {
    int i = blockIdx.x * blockDim.x + threadIdx.x;
    if (i < n) {
        float d = deg[i];
        dinv[i] = (d > 0.0f) ? rsqrtf(d) : 0.0f;
    }
}

__global__ void gcn_edge_norm(const int* __restrict__ ei, const float* __restrict__ dinv,
                              float* __restrict__ norm, int e) {
    int i = blockIdx.x * blockDim.x + threadIdx.x;
    if (i < e) norm[i] = dinv[ei[i]] * dinv[ei[GCN_E + i]];
}

// ---------------- WMMA GEMM: out[N x 64] = act(in)[N x DIN] @ W[DIN x 64] --
// One wave computes one 16-row slab across all 64 output columns (4 tiles of
// 16x16), stepping K by 4 with V_WMMA_F32_16X16X4_F32.
// A-fragment (32-bit A 16x4): lane l: M=l&15, K-pair = (l>>4)*2  -> one b64.
// B-fragment mirrors it: lane l: N=l&15, K-pair = (l>>4)*2.
// W is staged in LDS k-paired: Wl[(k/2)*PAIR_STRIDE + n*2 + (k&1)], so each
// B fragment is one aligned ds_load_b64 straight into an even VGPR pair.
// C layout: vgpr r, lane l -> row = r + 8*(l>>4), col = l&15.

template <int DIN, bool RELU>
__global__ __launch_bounds__(256) void gcn_gemm_wmma(const float* __restrict__ in,
                                                     const float* __restrict__ W,
                                                     float* __restrict__ out) {
    __shared__ float Wl[(DIN / 2) * PAIR_STRIDE];
    const int tid = threadIdx.x;
    for (int i = tid; i < DIN * 64; i += 256) {
        int k = i >> 6;          // 0..DIN-1
        int n = i & 63;          // 0..63
        Wl[(k >> 1) * PAIR_STRIDE + n * 2 + (k & 1)] = W[i];
    }
    __syncthreads();

    const int lane = tid & 31;
    const int wave = tid >> 5;
    const int tile = blockIdx.x * 8 + wave;          // 16-row tile index
    if (tile >= (GCN_N >> 4)) return;                // wave-uniform guard

    const int m    = lane & 15;
    const int hi   = lane >> 4;
    const int koff = hi * 2;                         // K-pair offset: 0 or 2
    const float* arow = in + (size_t)(tile * 16 + m) * DIN;
    const float* brow = &Wl[(koff >> 1) * PAIR_STRIDE + m * 2];

    v8f acc0 = {}, acc1 = {}, acc2 = {}, acc3 = {};
    for (int k0 = 0; k0 < DIN; k0 += 4) {
        float2 av = *(const float2*)(arow + k0 + koff);
        if (RELU) { av.x = fmaxf(av.x, 0.0f); av.y = fmaxf(av.y, 0.0f); }
        v2f a; a.x = av.x; a.y = av.y;

        const float* bp = brow + (k0 >> 1) * PAIR_STRIDE;
        v2f b;
        b = *(const v2f*)(bp +  0);
        acc0 = __builtin_amdgcn_wmma_f32_16x16x4_f32(false, a, false, b, (short)0, acc0, false, false);
        b = *(const v2f*)(bp + 32);
        acc1 = __builtin_amdgcn_wmma_f32_16x16x4_f32(false, a, false, b, (short)0, acc1, false, false);
        b = *(const v2f*)(bp + 64);
        acc2 = __builtin_amdgcn_wmma_f32_16x16x4_f32(false, a, false, b, (short)0, acc2, false, false);
        b = *(const v2f*)(bp + 96);
        acc3 = __builtin_amdgcn_wmma_f32_16x16x4_f32(false, a, false, b, (short)0, acc3, false, false);
    }

    float* orow = out + (size_t)(tile * 16 + 8 * hi) * 64 + m;
#pragma unroll
    for (int r = 0; r < 8; ++r) {
        orow[(size_t)r * 64 +  0] = acc0[r];
        orow[(size_t)r * 64 + 16] = acc1[r];
        orow[(size_t)r * 64 + 32] = acc2[r];
        orow[(size_t)r * 64 + 48] = acc3[r];
    }
}

// ---------------- aggregation: bias + self-loop init -----------------------
// out[i][:] = b[:] + (2*dinv[i]^2) * t[i][:]   (16 threads/node, float4 each)

__global__ void gcn_agg_init(const float* __restrict__ t, const float* __restrict__ dinv,
                             const float* __restrict__ bias, float* __restrict__ out) {
    int idx = blockIdx.x * blockDim.x + threadIdx.x;
    if (idx >= GCN_N * 16) return;
    int i = idx >> 4;
    int q = (idx & 15) * 4;
    float di = dinv[i];
    float s  = 2.0f * di * di;
    float4 tv = *(const float4*)(t + (size_t)i * 64 + q);
    float4 o;
    o.x = bias[q + 0] + s * tv.x;
    o.y = bias[q + 1] + s * tv.y;
    o.z = bias[q + 2] + s * tv.z;
    o.w = bias[q + 3] + s * tv.w;
    *(float4*)(out + (size_t)i * 64 + q) = o;
}

// ---------------- aggregation: edge scatter (atomic) -----------------------
// 16 threads per edge, each covers 4 features: out[dst] += norm_e * t[src].

__global__ void gcn_scatter(const int* __restrict__ ei, const float* __restrict__ norm,
                            const float* __restrict__ t, float* __restrict__ out) {
    int idx = blockIdx.x * blockDim.x + threadIdx.x;
    if (idx >= GCN_E * 16) return;
    int e = idx >> 4;
    int q = (idx & 15) * 4;
    int s = ei[e];
    int d = ei[GCN_E + e];
    float w = norm[e];
    float4 tv = *(const float4*)(t + (size_t)s * 64 + q);
    float* o = out + (size_t)d * 64 + q;
    atomicAdd(o + 0, w * tv.x);
    atomicAdd(o + 1, w * tv.y);
    atomicAdd(o + 2, w * tv.z);
    atomicAdd(o + 3, w * tv.w);
}

// ---------------------------------------------------------------------------

extern "C" void kernel_launch(void* const* d_in, const int* in_sizes, int n_in,
                              void* d_out, int out_size, void* d_ws, size_t ws_size,
                              hipStream_t stream) {
    (void)in_sizes; (void)n_in; (void)out_size; (void)ws_size;

    const float* x  = (const float*)d_in[0];
    const int*   ei = (const int*)d_in[1];
    const float* W[5] = {(const float*)d_in[2], (const float*)d_in[4], (const float*)d_in[6],
                         (const float*)d_in[8], (const float*)d_in[10]};
    const float* B[5] = {(const float*)d_in[3], (const float*)d_in[5], (const float*)d_in[7],
                         (const float*)d_in[9], (const float*)d_in[11]};

    // workspace carve-up (256B aligned)
    char*  ws  = (char*)d_ws;
    size_t off = 0;
    auto carve = [&](size_t bytes) {
        void* p = ws + off;
        off = (off + bytes + 255) & ~(size_t)255;
        return p;
    };
    float* deg  = (float*)carve((size_t)GCN_N * 4);
    float* dinv = (float*)carve((size_t)GCN_N * 4);
    float* nrm  = (float*)carve((size_t)GCN_E * 4);
    float* t    = (float*)carve((size_t)GCN_N * 64 * 4);
    float* hagg = (float*)carve((size_t)GCN_N * 64 * 4);

    const int BN  = (GCN_N + 255) / 256;
    const int BE  = (GCN_E + 255) / 256;
    const int BNF = (GCN_N * 16 + 255) / 256;   // node * 16 lanes
    const int BEF = (GCN_E * 16 + 255) / 256;   // edge * 16 lanes
    const int BG  = ((GCN_N / 16) + 7) / 8;     // 3125 tiles / 8 waves per block

    // --- normalization precompute ---
    gcn_deg_init <<<BN, 256, 0, stream>>>(deg, GCN_N);
    gcn_deg_edges<<<BE, 256, 0, stream>>>(ei, deg, GCN_E);
    gcn_dinv     <<<BN, 256, 0, stream>>>(deg, dinv, GCN_N);
    gcn_edge_norm<<<BE, 256, 0, stream>>>(ei, dinv, nrm, GCN_E);

    // --- layer 1: x[N,128] @ W1 ---
    gcn_gemm_wmma<GCN_IN, false><<<BG, 256, 0, stream>>>(x, W[0], t);
    gcn_agg_init<<<BNF, 256, 0, stream>>>(t, dinv, B[0], hagg);
    gcn_scatter <<<BEF, 256, 0, stream>>>(ei, nrm, t, hagg);

    // --- layers 2..4: relu(hagg) @ W ---
    for (int L = 1; L <= 3; ++L) {
        gcn_gemm_wmma<GCN_H, true><<<BG, 256, 0, stream>>>(hagg, W[L], t);
        gcn_agg_init<<<BNF, 256, 0, stream>>>(t, dinv, B[L], hagg);
        gcn_scatter <<<BEF, 256, 0, stream>>>(ei, nrm, t, hagg);
    }

    // --- layer 5: relu(hagg) @ W5 -> d_out (no final relu) ---
    float* out = (float*)d_out;
    gcn_gemm_wmma<GCN_H, true><<<BG, 256, 0, stream>>>(hagg, W[4], t);
    gcn_agg_init<<<BNF, 256, 0, stream>>>(t, dinv, B[4], out);
    gcn_scatter <<<BEF, 256, 0, stream>>>(ei, nrm, t, out);
}